// SoftDTW_44942537785910
// MI455X (gfx1250) — compile-verified
//
#include <hip/hip_runtime.h>
#include <hip/hip_bf16.h>

// Problem constants (from reference: N=M=4096, d=64, gamma=1, BIG=1e8)
#define NN 4096
#define MM 4096
#define DD 64
#define BIGV 1.0e8f

typedef __attribute__((ext_vector_type(2))) float v2f;
typedef __attribute__((ext_vector_type(8))) float v8f;

// ---------------------------------------------------------------------------
// Kernel 0: row norms  x2[i] = sum_k x[i][k]^2 ;  y2[j] = sum_k y[j][k]^2
// ---------------------------------------------------------------------------
__global__ void sdtw_norms_kernel(const float* __restrict__ x,
                                  const float* __restrict__ y,
                                  float* __restrict__ x2,
                                  float* __restrict__ y2) {
    int t = blockIdx.x * blockDim.x + threadIdx.x;
    const float* src;
    float* dst;
    int row;
    if (t < NN) {
        src = x; dst = x2; row = t;
    } else if (t < NN + MM) {
        src = y; dst = y2; row = t - NN;
    } else {
        return;
    }
    float s = 0.0f;
#pragma unroll 8
    for (int k = 0; k < DD; ++k) {
        float v = src[row * DD + k];
        s = fmaf(v, v, s);
    }
    dst[row] = s;
}

// ---------------------------------------------------------------------------
// Kernel 1: D = x2 + y2 - 2 * x @ y^T  via V_WMMA_F32_16X16X4_F32
// One wave computes one 16x16 tile of D; 8 waves (256 threads) per block.
//
// fp32 16x16x4 WMMA VGPR layouts (ISA 7.12.2):
//   A (16x4, MxK):  lanes 0-15 -> M=lane, {v0,v1} = K={0,1};
//                   lanes 16-31 -> M=lane-16, {v0,v1} = K={2,3}
//   B (4x16, KxN):  mirrored: lanes 0-15 -> N=lane, {v0,v1} = K={0,1};
//                   lanes 16-31 -> N=lane-16, {v0,v1} = K={2,3}
//   C/D (16x16):    vgpr r, lanes 0-15 -> M=r, N=lane;
//                   lanes 16-31 -> M=r+8, N=lane-16
// ---------------------------------------------------------------------------
__global__ __launch_bounds__(256)
void sdtw_dist_wmma_kernel(const float* __restrict__ x,
                           const float* __restrict__ y,
                           const float* __restrict__ x2,
                           const float* __restrict__ y2,
                           float* __restrict__ D) {
    const int lane = threadIdx.x & 31;
    const int wave = threadIdx.x >> 5;
    const int half = lane >> 4;   // 0: low 16 lanes, 1: high 16 lanes
    const int l16  = lane & 15;

    const int tile = blockIdx.x * 8 + wave;     // 65536 tiles total
    const int tm = tile >> 8;                   // 0..255  (row-tile)
    const int tn = tile & 255;                  // 0..255  (col-tile)

    const int arow = tm * 16 + l16;             // A row  (x row)
    const int bcol = tn * 16 + l16;             // B col  (y row)

    v8f acc = {};                               // fp32 accumulator, 8 VGPRs

#if __has_builtin(__builtin_amdgcn_wmma_f32_16x16x4_f32)
#pragma unroll
    for (int kk = 0; kk < DD; kk += 4) {
        const int ka = kk + half * 2;           // this half-wave's K pair
        v2f a, b;
        a.x = x[arow * DD + ka];
        a.y = x[arow * DD + ka + 1];
        b.x = y[bcol * DD + ka];
        b.y = y[bcol * DD + ka + 1];
        // 8 args: (neg_a, A, neg_b, B, c_mod, C, reuse_a, reuse_b)
        acc = __builtin_amdgcn_wmma_f32_16x16x4_f32(
            false, a, false, b, (short)0, acc, false, false);
    }
#else
    // Scalar fallback (should not be taken on gfx1250): plain dot per (m,n).
#pragma unroll
    for (int r = 0; r < 8; ++r) {
        const int gm = tm * 16 + r + half * 8;
        float s = 0.0f;
        for (int k = 0; k < DD; ++k)
            s = fmaf(x[gm * DD + k], y[bcol * DD + k], s);
        acc[r] = s;
    }
#endif

    // Epilogue: D[m][n] = x2[m] + y2[n] - 2*dot
    const float y2c = y2[bcol];
#pragma unroll
    for (int r = 0; r < 8; ++r) {
        const int gm = tm * 16 + r + half * 8;  // global row of this element
        const float val = x2[gm] + y2c - 2.0f * acc[r];
        D[gm * MM + bcol] = val;
    }
}

// ---------------------------------------------------------------------------
// Kernel 2: soft-DTW anti-diagonal wavefront. Single workgroup (1024 thr =
// 32 wave32 waves on one WGP). Three ping-pong diagonal buffers in LDS
// (3 x 4098 floats ~= 48 KB of the 320 KB WGP LDS). Sequential depth is
// N+M-1 = 8191 block barriers -- far cheaper than any cross-WGP sync.
// ---------------------------------------------------------------------------
__global__ __launch_bounds__(1024)
void sdtw_wavefront_kernel(const float* __restrict__ D,
                           float* __restrict__ out) {
    __shared__ float sbuf[3][NN + 2];
    const int tid = threadIdx.x;

    // init: k=0 diag -> buf0 (R[0,0]=0, rest BIG); k=1 diag -> buf1 (all BIG)
    for (int idx = tid; idx < NN + 2; idx += 1024) {
        sbuf[0][idx] = (idx == 0) ? 0.0f : BIGV;
        sbuf[1][idx] = BIGV;
        sbuf[2][idx] = BIGV;
    }
    __syncthreads();

    int cur = 2;                         // buffer holding diagonal k
    for (int k = 2; k <= NN + MM; ++k) {
        const int im1 = (cur + 2) % 3;   // k-1
        const int im2 = (cur + 1) % 3;   // k-2
        float* __restrict__ rk   = sbuf[cur];
        const float* __restrict__ rkm1 = sbuf[im1];
        const float* __restrict__ rkm2 = sbuf[im2];

        for (int i = 1 + tid; i <= NN; i += 1024) {
            const int j = k - i;
            float val = BIGV;
            if (j >= 1 && j <= MM) {
                // gamma == 1: softmin = -(log(sum exp(rX - rmax)) + rmax)
                const float r0 = -rkm2[i - 1];     // R[i-1, j-1]
                const float r1 = -rkm1[i - 1];     // R[i-1, j  ]
                const float r2 = -rkm1[i];         // R[i,   j-1]
                const float rmax = fmaxf(fmaxf(r0, r1), r2);
                const float rsum = __expf(r0 - rmax) +
                                   __expf(r1 - rmax) +
                                   __expf(r2 - rmax);
                const float softmin = -(__logf(rsum) + rmax);
                val = D[(i - 1) * MM + (j - 1)] + softmin;
                // warm L2/L0 for next diagonal's D element for this row
                if (j < MM)
                    __builtin_prefetch(&D[(i - 1) * MM + j], 0, 1);
            }
            rk[i] = val;
        }
        if (tid == 0) rk[0] = BIGV;      // keep boundary BIG after reuse
        __syncthreads();
        cur = cur + 1;
        if (cur == 3) cur = 0;
    }

    // last written diagonal (k = N+M) lives in (cur+2)%3; R[N,M] at index N
    if (tid == 0) {
        const float rNM = sbuf[(cur + 2) % 3][NN];
        out[0] = rNM * (1.0f / (float)(NN + MM));
    }
}

// ---------------------------------------------------------------------------
// Launch: workspace layout  [ x2 : 4096 f32 | y2 : 4096 f32 | D : 4096x4096 ]
// ---------------------------------------------------------------------------
extern "C" void kernel_launch(void* const* d_in, const int* in_sizes, int n_in,
                              void* d_out, int out_size, void* d_ws, size_t ws_size,
                              hipStream_t stream) {
    (void)in_sizes; (void)n_in; (void)out_size; (void)ws_size;

    const float* x = (const float*)d_in[0];
    const float* y = (const float*)d_in[1];
    float* out = (float*)d_out;

    float* x2 = (float*)d_ws;
    float* y2 = x2 + NN;
    float* D  = y2 + MM;   // 64 MB, L2-resident on MI455X (192 MB L2)

    // 1) row norms
    sdtw_norms_kernel<<<(NN + MM + 255) / 256, 256, 0, stream>>>(x, y, x2, y2);

    // 2) distance matrix via fp32 WMMA: 256x256 tiles, 8 tiles/block
    sdtw_dist_wmma_kernel<<<(256 * 256) / 8, 256, 0, stream>>>(x, y, x2, y2, D);

    // 3) soft-DTW wavefront on one WGP
    sdtw_wavefront_kernel<<<1, 1024, 0, stream>>>(D, out);
}